// MessageFactoryBayonet2Helium_1228360646896
// MI455X (gfx1250) — compile-verified
//
#include <hip/hip_runtime.h>

#define PI_F 3.14159265358979323846f

#define WAVE_SIZE        32
#define WAVES_PER_BLOCK  8
#define BLOCK_SIZE       (WAVE_SIZE * WAVES_PER_BLOCK)   // 256
#define TILE_EDGES       (WAVE_SIZE * 4)                 // 128 edges per wave-tile (16B/lane/array)
#define MAX_BLOCKS       2048

// Exact types the gfx1250 async-to-LDS builtin expects: pointer-to-int4 (clang
// vector type) in AS1 (global) and AS3 (LDS).
typedef int v4i __attribute__((ext_vector_type(4)));
typedef __attribute__((address_space(1))) v4i GV4;
typedef __attribute__((address_space(3))) v4i LV4;

#if __has_builtin(__builtin_amdgcn_global_load_async_to_lds_b128)
#define HAVE_ASYNC_LDS 1
#else
#define HAVE_ASYNC_LDS 0
#endif

// cpol: TH[2:0]=1 -> non-temporal (edge stream is read exactly once).
#define ASYNC_CPOL 1

// 16-byte global -> LDS copy: async data-mover path on CDNA5, plain copy otherwise.
__device__ __forceinline__ void copy16_g2l(const void* g, void* l) {
#if HAVE_ASYNC_LDS
    __builtin_amdgcn_global_load_async_to_lds_b128((GV4*)g, (LV4*)l, 0, ASYNC_CPOL);
#else
    *(int4*)l = *(const int4*)g;
#endif
}

template <int N>
__device__ __forceinline__ void wait_async() {
#if HAVE_ASYNC_LDS
#if __has_builtin(__builtin_amdgcn_s_wait_asynccnt)
    __builtin_amdgcn_s_wait_asynccnt(N);
#else
    asm volatile("s_wait_asynccnt %0" ::"n"(N) : "memory");
#endif
#endif
}

// Scatter-add: force the native no-return global_atomic_add_f32 (L2 atomic
// units) instead of a CAS retry loop by declaring no fine-grained/remote
// memory and denormal-mode indifference.
__device__ __forceinline__ void scatter_add(float* __restrict__ p, float v) {
#if defined(__has_cpp_attribute) && __has_cpp_attribute(clang::atomic)
    [[clang::atomic(no_remote_memory, no_fine_grained_memory, ignore_denormal_mode)]] {
        __hip_atomic_fetch_add(p, v, __ATOMIC_RELAXED, __HIP_MEMORY_SCOPE_AGENT);
    }
#else
    __hip_atomic_fetch_add(p, v, __ATOMIC_RELAXED, __HIP_MEMORY_SCOPE_AGENT);
#endif
}

__device__ __forceinline__ void edge_compute(
    int s, int g, float c, float dt,
    const float* __restrict__ T, const float* __restrict__ L,
    const float* __restrict__ D, const float* __restrict__ A,
    float* __restrict__ out)
{
    float Ts = T[s];
    float Tt = T[g];
    float d  = fmaxf(Ts - Tt, 0.0f);          // relu(T_src - T_tgt)
    float wet = L[s] * D[s] * PI_F * A[s];    // wetted area
    float val = d * c * 0.5f * wet * (Tt * Tt * Tt) * dt;
    if (val != 0.0f) {                        // ~50% of edges contribute 0; skip atomic
        scatter_add(&out[g], val);
    }
}

__global__ __launch_bounds__(BLOCK_SIZE)
void kapitza_edge_kernel(const float* __restrict__ T,
                         const float* __restrict__ L,
                         const float* __restrict__ D,
                         const float* __restrict__ A,
                         const float* __restrict__ cond,
                         const int*   __restrict__ src,
                         const int*   __restrict__ tgt,
                         const float* __restrict__ dt_ptr,
                         float*       __restrict__ out,
                         int nEdges)
{
    // Per-wave double-buffered staging: [wave][buf][128]
    __shared__ alignas(16) int   sh_src[WAVES_PER_BLOCK][2][TILE_EDGES];
    __shared__ alignas(16) int   sh_tgt[WAVES_PER_BLOCK][2][TILE_EDGES];
    __shared__ alignas(16) float sh_cnd[WAVES_PER_BLOCK][2][TILE_EDGES];

    const float dt   = *dt_ptr;
    const int   lane = threadIdx.x & (WAVE_SIZE - 1);
    const int   wib  = threadIdx.x >> 5;
    const int   nwaves = (int)gridDim.x * WAVES_PER_BLOCK;
    const int   wgid   = (int)blockIdx.x * WAVES_PER_BLOCK + wib;
    const int   ntiles = nEdges / TILE_EDGES;

    auto stage = [&](int t, int b) {
        const long e0 = (long)t * TILE_EDGES;
        const int  q  = lane * 4;               // this lane's 16B chunk
        copy16_g2l(src  + e0 + q, &sh_src[wib][b][q]);
        copy16_g2l(tgt  + e0 + q, &sh_tgt[wib][b][q]);
        copy16_g2l(cond + e0 + q, &sh_cnd[wib][b][q]);
    };

    int buf = 0;
    int t   = wgid;
    if (t < ntiles) stage(t, 0);

    for (; t < ntiles; t += nwaves) {
        const int tn = t + nwaves;
        if (tn < ntiles) {
            stage(tn, buf ^ 1);   // prefetch next tile (async, in-flight)
            wait_async<3>();      // in-order completion: oldest 3 done => tile t resident
        } else {
            wait_async<0>();
        }

        const int4   s4 = *(const int4*)  &sh_src[wib][buf][lane * 4];
        const int4   g4 = *(const int4*)  &sh_tgt[wib][buf][lane * 4];
        const float4 c4 = *(const float4*)&sh_cnd[wib][buf][lane * 4];

        edge_compute(s4.x, g4.x, c4.x, dt, T, L, D, A, out);
        edge_compute(s4.y, g4.y, c4.y, dt, T, L, D, A, out);
        edge_compute(s4.z, g4.z, c4.z, dt, T, L, D, A, out);
        edge_compute(s4.w, g4.w, c4.w, dt, T, L, D, A, out);

        buf ^= 1;
    }

    // Tail edges (nEdges % 128), direct path.
    const long gtid    = (long)blockIdx.x * BLOCK_SIZE + threadIdx.x;
    const long gstride = (long)gridDim.x * BLOCK_SIZE;
    for (long e = (long)ntiles * TILE_EDGES + gtid; e < nEdges; e += gstride) {
        edge_compute(src[e], tgt[e], cond[e], dt, T, L, D, A, out);
    }
}

__global__ __launch_bounds__(BLOCK_SIZE)
void zero_f32_kernel(float* __restrict__ out, int n)
{
    const int i4 = (blockIdx.x * BLOCK_SIZE + threadIdx.x) * 4;
    if (i4 + 4 <= n) {
        *(float4*)(out + i4) = make_float4(0.f, 0.f, 0.f, 0.f);
    } else {
        for (int k = i4; k < n; ++k) out[k] = 0.f;
    }
}

extern "C" void kernel_launch(void* const* d_in, const int* in_sizes, int n_in,
                              void* d_out, int out_size, void* d_ws, size_t ws_size,
                              hipStream_t stream)
{
    (void)n_in; (void)d_ws; (void)ws_size;
    const float* T    = (const float*)d_in[0];
    const float* L    = (const float*)d_in[1];
    const float* D    = (const float*)d_in[2];
    const float* A    = (const float*)d_in[3];
    const float* cond = (const float*)d_in[4];
    const int*   src  = (const int*)  d_in[5];
    const int*   tgt  = (const int*)  d_in[6];
    const float* dt   = (const float*)d_in[7];
    float*       out  = (float*)d_out;

    const int nEdges = in_sizes[4];
    const int nNodes = out_size;

    // 1) zero the pooled output (harness poisons d_out)
    const int zthreads = (nNodes + 3) / 4;
    const int zblocks  = (zthreads + BLOCK_SIZE - 1) / BLOCK_SIZE;
    zero_f32_kernel<<<zblocks > 0 ? zblocks : 1, BLOCK_SIZE, 0, stream>>>(out, nNodes);

    // 2) edge gather/compute/scatter
    const long ntiles = nEdges / TILE_EDGES;
    long blocks = (ntiles + WAVES_PER_BLOCK - 1) / WAVES_PER_BLOCK;
    if (blocks > MAX_BLOCKS) blocks = MAX_BLOCKS;
    if (blocks < 1) blocks = 1;
    kapitza_edge_kernel<<<(int)blocks, BLOCK_SIZE, 0, stream>>>(
        T, L, D, A, cond, src, tgt, dt, out, nEdges);
}